// MoLELayer_57690000720299
// MI455X (gfx1250) — compile-verified
//
#include <hip/hip_runtime.h>
#include <cmath>

#define BB 8
#define LL 4096
#define DD 4096
#define EE 8
#define RR 16
#define KK 2
#define EPSV 1e-5f
#define SCALEV (1.0f / 16.0f)   // ALPHA / R

typedef __attribute__((ext_vector_type(2))) float v2f;
typedef __attribute__((ext_vector_type(8))) float v8f;

__device__ __forceinline__ float wave_sum32(float v) {
#pragma unroll
    for (int o = 16; o > 0; o >>= 1) v += __shfl_down(v, o, 32);
    return v;
}

// ---------------------------------------------------------------------------
// K1: partial column sums of x over 256-row L-chunks.
// grid = B(8) * Dchunks(4) * Lchunks(16) = 512 blocks, 256 threads, float4.
// hpart layout: [16][B][D]
// ---------------------------------------------------------------------------
__global__ void k1_partial_mean(const float* __restrict__ x, float* __restrict__ hpart) {
    int bx = blockIdx.x;
    int lchunk = bx & 15;
    int dchunk = (bx >> 4) & 3;
    int b = bx >> 6;
    int d = dchunk * 1024 + threadIdx.x * 4;

    const float4* xp = (const float4*)x;
    size_t idx4 = (((size_t)b * LL + (size_t)lchunk * 256) * DD + d) >> 2;
    float4 acc = {0.f, 0.f, 0.f, 0.f};
    for (int l = 0; l < 256; ++l) {
        float4 v = xp[idx4];
        acc.x += v.x; acc.y += v.y; acc.z += v.z; acc.w += v.w;
        idx4 += DD / 4;
    }
    ((float4*)hpart)[(((size_t)lchunk * BB + b) * DD + d) >> 2] = acc;
}

// ---------------------------------------------------------------------------
// K1b: deterministic reduce of 16 partials -> h = mean over L.  8192 threads.
// ---------------------------------------------------------------------------
__global__ void k1b_reduce_mean(const float* __restrict__ hpart, float* __restrict__ h) {
    int gid = blockIdx.x * blockDim.x + threadIdx.x;   // 0 .. B*D/4-1
    const float4* hp = (const float4*)hpart;
    float4 acc = {0.f, 0.f, 0.f, 0.f};
#pragma unroll
    for (int p = 0; p < 16; ++p) {
        float4 v = hp[(size_t)p * (BB * DD / 4) + gid];
        acc.x += v.x; acc.y += v.y; acc.z += v.z; acc.w += v.w;
    }
    const float inv = 1.0f / (float)LL;
    acc.x *= inv; acc.y *= inv; acc.z *= inv; acc.w *= inv;
    ((float4*)h)[gid] = acc;
}

// ---------------------------------------------------------------------------
// K2a: router. grid = B, 256 threads (8 waves). Wave e computes logits[e]
// via wave32 shuffle reduce; thread 0 does top-2 + softmax.
// ---------------------------------------------------------------------------
__global__ void k2a_router(const float* __restrict__ h, const float* __restrict__ gate_w,
                           const float* __restrict__ gate_b,
                           float* __restrict__ wgt, int* __restrict__ eidx) {
    int b = blockIdx.x;
    int wid = threadIdx.x >> 5, lane = threadIdx.x & 31;
    const float* gw = gate_w + (size_t)wid * DD;
    const float* hb = h + (size_t)b * DD;
    float acc = 0.f;
    for (int d = lane; d < DD; d += 32) acc += gw[d] * hb[d];
    acc = wave_sum32(acc);

    __shared__ float lg[EE];
    if (lane == 0) lg[wid] = acc + gate_b[wid];
    __syncthreads();

    if (threadIdx.x == 0) {
        int i0 = 0; float v0 = lg[0];
        for (int e = 1; e < EE; ++e) if (lg[e] > v0) { v0 = lg[e]; i0 = e; }
        int i1 = (i0 == 0) ? 1 : 0; float v1 = lg[i1];
        for (int e = 0; e < EE; ++e)
            if (e != i0 && lg[e] > v1) { v1 = lg[e]; i1 = e; }
        float e1 = expf(v1 - v0);          // v0 >= v1
        float s = 1.0f + e1;
        wgt[b * 2 + 0] = 1.0f / s;
        wgt[b * 2 + 1] = e1 / s;
        eidx[b * 2 + 0] = i0;
        eidx[b * 2 + 1] = i1;
    }
}

// ---------------------------------------------------------------------------
// K2b: z[k,r] = sum_d A_w[e_k,r,d] * h[d] via V_WMMA_F32_16X16X4_F32.
// grid = B*K blocks, 128 threads (4 waves); wave w covers D range [w*1024,+1024).
// fp32 A tile 16x4: lane holds M=lane&15; VGPR v holds K = 2*(lane>=16)+v.
// B tile 4x16 broadcasts h into every column (same lane-half/VGPR K map),
// so D columns are identical and z is read from lanes 0 / 16.
// Each wave writes its own zpart slot -> deterministic, no atomics.
// ---------------------------------------------------------------------------
__global__ void k2b_z_wmma(const float* __restrict__ A_w, const float* __restrict__ h,
                           const int* __restrict__ eidx, float* __restrict__ zpart) {
    int bk = blockIdx.x;           // b*K + k
    int b = bk >> 1;
    int wid = threadIdx.x >> 5, lane = threadIdx.x & 31;
    int e = eidx[bk];
    int m = lane & 15;
    int khalf = (lane >> 4) * 2;

    const float* Arow = A_w + ((size_t)e * RR + m) * DD;
    const float* hb = h + (size_t)b * DD;

    v8f c = {};
    int d0 = wid * 1024;
    int dend = d0 + 1024;
    for (; d0 < dend; d0 += 4) {
        v2f a, bb;
        a.x = Arow[d0 + khalf];
        a.y = Arow[d0 + khalf + 1];
        bb.x = hb[d0 + khalf];
        bb.y = hb[d0 + khalf + 1];
        c = __builtin_amdgcn_wmma_f32_16x16x4_f32(false, a, false, bb,
                                                  (short)0, c, false, false);
    }

    float* zp = zpart + ((size_t)bk * 4 + wid) * RR;
    if (lane == 0) {
#pragma unroll
        for (int r = 0; r < 8; ++r) zp[r] = c[r];
    }
    if (lane == 16) {
#pragma unroll
        for (int r = 0; r < 8; ++r) zp[8 + r] = c[r];
    }
}

// ---------------------------------------------------------------------------
// K2c: delta[b,d] = scale * sum_k w_k * sum_r B_w[e_k,d,r] * z[b,k,r]
// grid = B*16 blocks, 256 threads: one thread per (b,d).
// ---------------------------------------------------------------------------
__global__ void k2c_delta(const float* __restrict__ B_w, const float* __restrict__ zpart,
                          const float* __restrict__ wgt, const int* __restrict__ eidx,
                          float* __restrict__ delta) {
    int blk = blockIdx.x;
    int b = blk >> 4;
    int d = (blk & 15) * 256 + threadIdx.x;

    float acc = 0.f;
#pragma unroll
    for (int k = 0; k < KK; ++k) {
        int e = eidx[b * 2 + k];
        float wk = wgt[b * 2 + k];
        const float* zp = zpart + ((size_t)(b * 2 + k)) * 4 * RR;
        const float* Brow = B_w + ((size_t)e * DD + d) * RR;
        float part = 0.f;
#pragma unroll
        for (int r = 0; r < RR; ++r) {
            float zr = zp[r] + zp[RR + r] + zp[2 * RR + r] + zp[3 * RR + r];
            part += Brow[r] * zr;
        }
        acc += wk * part;
    }
    delta[(size_t)b * DD + d] = acc * SCALEV;
}

// ---------------------------------------------------------------------------
// K3: out = LayerNorm(x + delta) per row. grid = B*L blocks, 256 threads.
// 4 x float4 per thread kept in registers; two-level wave32 reduction.
// ---------------------------------------------------------------------------
__global__ void k3_layernorm(const float* __restrict__ x, const float* __restrict__ delta,
                             const float* __restrict__ gamma, const float* __restrict__ beta,
                             float* __restrict__ out) {
    int row = blockIdx.x;
    int b = row >> 12;                     // row / L
    size_t base = (size_t)row * DD;
    const float4* xp = (const float4*)(x + base);
    const float4* dp = (const float4*)(delta + (size_t)b * DD);
    int t = threadIdx.x;

    float4 y[4];
    float s = 0.f, ss = 0.f;
#pragma unroll
    for (int i = 0; i < 4; ++i) {
        int j = i * 256 + t;
        float4 xv = xp[j], dv = dp[j];
        float4 yv;
        yv.x = xv.x + dv.x; yv.y = xv.y + dv.y;
        yv.z = xv.z + dv.z; yv.w = xv.w + dv.w;
        s  += yv.x + yv.y + yv.z + yv.w;
        ss += yv.x * yv.x + yv.y * yv.y + yv.z * yv.z + yv.w * yv.w;
        y[i] = yv;
    }

    int wid = t >> 5, lane = t & 31;
    s = wave_sum32(s);
    ss = wave_sum32(ss);
    __shared__ float rs[8], rss[8], fin[2];
    if (lane == 0) { rs[wid] = s; rss[wid] = ss; }
    __syncthreads();
    if (t == 0) {
        float ts = 0.f, tss = 0.f;
#pragma unroll
        for (int i = 0; i < 8; ++i) { ts += rs[i]; tss += rss[i]; }
        float mu = ts * (1.0f / DD);
        float var = tss * (1.0f / DD) - mu * mu;   // biased, matches jnp.var
        fin[0] = mu;
        fin[1] = rsqrtf(var + EPSV);
    }
    __syncthreads();
    float mu = fin[0], rstd = fin[1];

    const float4* gp = (const float4*)gamma;
    const float4* bp = (const float4*)beta;
    float4* op = (float4*)(out + base);
#pragma unroll
    for (int i = 0; i < 4; ++i) {
        int j = i * 256 + t;
        float4 g = gp[j], be = bp[j], yv = y[i], o;
        o.x = (yv.x - mu) * rstd * g.x + be.x;
        o.y = (yv.y - mu) * rstd * g.y + be.y;
        o.z = (yv.z - mu) * rstd * g.z + be.z;
        o.w = (yv.w - mu) * rstd * g.w + be.w;
        op[j] = o;
    }
}

// ---------------------------------------------------------------------------
extern "C" void kernel_launch(void* const* d_in, const int* in_sizes, int n_in,
                              void* d_out, int out_size, void* d_ws, size_t ws_size,
                              hipStream_t stream) {
    const float* x      = (const float*)d_in[0];
    const float* gate_w = (const float*)d_in[1];
    const float* gate_b = (const float*)d_in[2];
    const float* A_w    = (const float*)d_in[3];
    const float* B_w    = (const float*)d_in[4];
    const float* gamma  = (const float*)d_in[5];
    const float* beta   = (const float*)d_in[6];
    float* out = (float*)d_out;

    // workspace layout (floats): ~2.36 MB total
    float* ws    = (float*)d_ws;
    float* hpart = ws;                          // 16*B*D
    float* h     = hpart + 16 * BB * DD;        // B*D
    float* delta = h + BB * DD;                 // B*D
    float* zpart = delta + BB * DD;             // B*K*4*R
    float* wgt   = zpart + BB * KK * 4 * RR;    // B*K
    int*   eidx  = (int*)(wgt + BB * KK);       // B*K

    k1_partial_mean<<<BB * 4 * 16, 256, 0, stream>>>(x, hpart);
    k1b_reduce_mean<<<(BB * DD / 4) / 256, 256, 0, stream>>>(hpart, h);
    k2a_router<<<BB, 256, 0, stream>>>(h, gate_w, gate_b, wgt, eidx);
    k2b_z_wmma<<<BB * KK, 128, 0, stream>>>(A_w, h, eidx, zpart);
    k2c_delta<<<BB * 16, 256, 0, stream>>>(B_w, zpart, wgt, eidx, delta);
    k3_layernorm<<<BB * LL, 256, 0, stream>>>(x, delta, gamma, beta, out);
}